// DigitCaps_34273839022736
// MI455X (gfx1250) — compile-verified
//
#include <hip/hip_runtime.h>
#include <math.h>

// Problem constants (match reference)
#define BATCH 32
#define PREV  16384
#define INCH  8
#define OUTCH 16
#define NCAPS 10

typedef __attribute__((ext_vector_type(2))) float v2f;
typedef __attribute__((ext_vector_type(8))) float v8f;

// ---------------------------------------------------------------------------
// Fused routing pass.
//   For each p, recompute u_hat[b, j, p, :] with V_WMMA_F32_16X16X4_F32
//   (M = o, N = b-tile of 16, K = d, two K=4 chunks), then do the routing
//   softmax over j lane-locally (D layout => lane holds one b, 8 o's) and
//   accumulate c * u_hat into register accumulators. Workgroup = 4 waves,
//   all on the same b-tile; waves cover disjoint p ranges; final reduction
//   goes LDS -> global_atomic_add_f32 into the tiny s buffer.
// ---------------------------------------------------------------------------
__global__ __launch_bounds__(128) void caps_route_pass(
    const float* __restrict__ x,     // (B, P, D)
    const float* __restrict__ W,     // (J, P, D, O)
    const float* __restrict__ Vacc,  // (B, J, O)  cumulative squashed v
    float* __restrict__ s_out)       // (B, J, O)  accumulated via atomics
{
  __shared__ float vacc_lds[16 * NCAPS * OUTCH];      // this block's b-slice (2560 f)
  __shared__ float sred[4 * NCAPS * 16 * OUTCH];      // per-wave partials (10240 f)

  const int bt     = blockIdx.x & 1;        // which b-tile (b 0-15 / 16-31)
  const int ptile  = blockIdx.x >> 1;
  const int tid    = threadIdx.x;
  const int wave   = tid >> 5;
  const int lane   = tid & 31;
  const int lh     = lane & 15;
  const int hi     = lane >> 4;             // 0: lanes 0-15, 1: lanes 16-31
  const int b_base = bt * 16;

  // Stage this b-tile's V_acc slice into LDS (contiguous [bi][j][o]).
  for (int i = tid; i < 16 * NCAPS * OUTCH; i += 128)
    vacc_lds[i] = Vacc[b_base * NCAPS * OUTCH + i];
  __syncthreads();

  v8f sacc[NCAPS];
  #pragma unroll
  for (int j = 0; j < NCAPS; ++j) { v8f z = {}; sacc[j] = z; }

  const int p0  = ptile * 128 + wave * 32;  // 32 p's per wave
  const int dlo = hi ? 2 : 0;               // K rows held by this lane half

  for (int pi = 0; pi < 32; ++pi) {
    const int p = p0 + pi;

    // ---- B matrix = x^T : B[k=d][n=b].  VGPR0 = K0/K2, VGPR1 = K1/K3 ----
    const float* xp = x + ((size_t)(b_base + lh) * PREV + p) * INCH + dlo;
    const float2 x01 = *(const float2*)(xp);      // d{0,1} (lo) / d{2,3} (hi)
    const float2 x45 = *(const float2*)(xp + 4);  // d{4,5} (lo) / d{6,7} (hi)
    v2f Bc0; Bc0[0] = x01.x; Bc0[1] = x01.y;      // K-chunk d0..3
    v2f Bc1; Bc1[0] = x45.x; Bc1[1] = x45.y;      // K-chunk d4..7

    float logit[NCAPS];
    v8f   u[NCAPS];

    #pragma unroll
    for (int j = 0; j < NCAPS; ++j) {
      // ---- A matrix = W^T : A[m=o][k=d] ----
      const float* wp = W + ((size_t)j * PREV + p) * (INCH * OUTCH) + dlo * OUTCH + lh;
      v2f A0; A0[0] = wp[0];         A0[1] = wp[OUTCH];      // d0/d2, d1/d3 @ o=lh
      v2f A1; A1[0] = wp[4 * OUTCH]; A1[1] = wp[5 * OUTCH];  // d4/d6, d5/d7

      v8f acc = {};
      acc = __builtin_amdgcn_wmma_f32_16x16x4_f32(false, A0, false, Bc0,
                                                  (short)0, acc, false, false);
      acc = __builtin_amdgcn_wmma_f32_16x16x4_f32(false, A1, false, Bc1,
                                                  (short)0, acc, false, false);
      u[j] = acc;   // lane: b = b_base+lh fixed, o = hi*8 + r across 8 VGPRs

      // logit[b,j] = u_hat . V_acc  (8 lane-local FMAs + one half-swap)
      const float* va = &vacc_lds[(lh * NCAPS + j) * OUTCH + hi * 8];
      const float4 va0 = *(const float4*)(va);
      const float4 va1 = *(const float4*)(va + 4);
      float ls = acc[0] * va0.x + acc[1] * va0.y + acc[2] * va0.z + acc[3] * va0.w
               + acc[4] * va1.x + acc[5] * va1.y + acc[6] * va1.z + acc[7] * va1.w;
      ls += __shfl_xor(ls, 16, 32);   // combine the two o-halves
      logit[j] = ls;
    }

    // ---- softmax over j, fully lane-local (J=10 values in registers) ----
    float m = logit[0];
    #pragma unroll
    for (int j = 1; j < NCAPS; ++j) m = fmaxf(m, logit[j]);
    float esum = 0.f;
    #pragma unroll
    for (int j = 0; j < NCAPS; ++j) {
      logit[j] = __expf(logit[j] - m);
      esum += logit[j];
    }
    const float inv = 1.0f / esum;

    #pragma unroll
    for (int j = 0; j < NCAPS; ++j) {
      const float c = logit[j] * inv;
      #pragma unroll
      for (int r = 0; r < 8; ++r) sacc[j][r] += c * u[j][r];
    }
  }

  // ---- reduce the 4 waves' partials through LDS, then f32 atomics ----
  #pragma unroll
  for (int j = 0; j < NCAPS; ++j) {
    float* dst = &sred[((wave * NCAPS + j) * 16 + lh) * OUTCH + hi * 8];
    *(float4*)(dst)     = make_float4(sacc[j][0], sacc[j][1], sacc[j][2], sacc[j][3]);
    *(float4*)(dst + 4) = make_float4(sacc[j][4], sacc[j][5], sacc[j][6], sacc[j][7]);
  }
  __syncthreads();

  for (int i = tid; i < NCAPS * 16 * OUTCH; i += 128) {
    const float v = sred[i] + sred[2560 + i] + sred[2 * 2560 + i] + sred[3 * 2560 + i];
    const int j  = i >> 8;          // layout [j][bi][o]
    const int bi = (i >> 4) & 15;
    const int o  = i & 15;
    atomicAdd(&s_out[(((size_t)(b_base + bi)) * NCAPS + j) * OUTCH + o], v);
  }
}

// ---------------------------------------------------------------------------
// v = squash(s); V_acc += v (or write v to output on the final pass).
// Also re-zeros s for the next pass. One thread per (b,j).
// ---------------------------------------------------------------------------
__global__ void caps_squash(float* __restrict__ s, float* __restrict__ Vacc,
                            float* __restrict__ out, int is_final)
{
  const int t = threadIdx.x;
  if (t >= BATCH * NCAPS) return;
  float sv[OUTCH];
  float sq = 0.f;
  #pragma unroll
  for (int o = 0; o < OUTCH; ++o) {
    sv[o] = s[t * OUTCH + o];
    sq += sv[o] * sv[o];
  }
  const float f = sq / ((1.f + sq) * sqrtf(sq));
  #pragma unroll
  for (int o = 0; o < OUTCH; ++o) {
    const float v = sv[o] * f;
    if (is_final) out[t * OUTCH + o] = v;
    else          Vacc[t * OUTCH + o] += v;
    s[t * OUTCH + o] = 0.f;
  }
}

__global__ void caps_zero(float* __restrict__ p, int n)
{
  const int i = blockIdx.x * blockDim.x + threadIdx.x;
  if (i < n) p[i] = 0.f;
}

// ---------------------------------------------------------------------------
extern "C" void kernel_launch(void* const* d_in, const int* in_sizes, int n_in,
                              void* d_out, int out_size, void* d_ws, size_t ws_size,
                              hipStream_t stream) {
  (void)in_sizes; (void)n_in; (void)out_size; (void)ws_size;
  const float* x = (const float*)d_in[0];   // (32, 16384, 8)
  const float* W = (const float*)d_in[1];   // (10, 16384, 8, 16)
  float* out  = (float*)d_out;              // (32, 10, 16) = 5120 floats
  float* s    = (float*)d_ws;                          // 5120 floats
  float* Vacc = s + BATCH * NCAPS * OUTCH;             // 5120 floats

  const int nz = 2 * BATCH * NCAPS * OUTCH;
  caps_zero<<<(nz + 255) / 256, 256, 0, stream>>>(s, nz);

  for (int it = 0; it < 3; ++it) {
    caps_route_pass<<<2 * (PREV / 128), 128, 0, stream>>>(x, W, Vacc, s);
    caps_squash<<<1, 320, 0, stream>>>(s, Vacc, out, it == 2);
  }
}